// SpikingCircadian_16673063043188
// MI455X (gfx1250) — compile-verified
//
#include <hip/hip_runtime.h>
#include <stdint.h>

// ---------------- problem constants (match reference) ----------------
#define N_POP              4194304
#define N_INNER            10
#define BLK                256
#define NPT                4                                // neurons per thread (main kernel)
#define MAIN_BLOCKS_PER_POP (N_POP / (BLK * NPT))           // 4096
#define A_BLOCKS_PER_POP   1024

// ws layout (floats)
#define WS_APART_OFF   0        // 2048 floats: input-rate partials (day 0..1023, night 1024..2047)
#define WS_SCAL_OFF    2048     // 8 floats: [0]=I_day_base [1]=I_night_base [2]=phase
#define WS_MPART_OFF   2064     // 8192 floats: final-rate partials (day 0..4095, night 4096..8191)

typedef float v2f __attribute__((ext_vector_type(2)));
typedef float v8f __attribute__((ext_vector_type(8)));

// ---- wave32 sum via V_WMMA_F32_16X16X4_F32 (A = ones 16x4, B rows 0/1 = lane values) ----
// D[m][n] = sum_k B[k][n] = val[lane n] + val[lane n+16]; finish 16->1 with 4 xor-shuffles.
__device__ __forceinline__ float wave_sum32(float val) {
  v2f a; a.x = 1.0f; a.y = 1.0f;       // 16x4 all-ones
  v2f b; b.x = val;  b.y = 0.0f;       // rows 0/1 hold the 32 lane values, rows 2/3 zero
  v8f c = {};
  c = __builtin_amdgcn_wmma_f32_16x16x4_f32(false, a, false, b, (short)0, c, false, false);
  float s = c[0];                      // column sums replicated over rows
  s += __shfl_xor(s, 1, 32);
  s += __shfl_xor(s, 2, 32);
  s += __shfl_xor(s, 4, 32);
  s += __shfl_xor(s, 8, 32);
  return s;                            // full-wave sum in every lane
}

// 256-thread block sum; result valid in thread 0. Safe to call repeatedly with same smem.
__device__ __forceinline__ float block_sum256(float v, float* smem) {
  float w = wave_sum32(v);
  int wid = threadIdx.x >> 5;
  if ((threadIdx.x & 31) == 0) smem[wid] = w;
  __syncthreads();
  float t = 0.0f;
  if (threadIdx.x == 0) {
#pragma unroll
    for (int i = 0; i < 8; ++i) t += smem[i];
  }
  __syncthreads();
  return t;
}

// ---- cheap deterministic counter RNG (good avalanche, i.i.d.-quality for 4M-neuron means) ----
__device__ __forceinline__ uint32_t lowbias32(uint32_t x) {
  x ^= x >> 16; x *= 0x7feb352du;
  x ^= x >> 15; x *= 0x846ca68bu;
  x ^= x >> 16;
  return x;
}

// Box-Muller pair scaled by NOISE_STD=0.3, keyed by a unique 32-bit counter.
__device__ __forceinline__ void noise_pair(uint32_t key, float* n0, float* n1) {
  uint32_t r1 = lowbias32(key ^ 0x85ebca6bu);
  uint32_t r2 = lowbias32(r1 + 0xc2b2ae35u);
  float u1 = (float)(r1 >> 8) * (1.0f / 16777216.0f) + (0.5f / 16777216.0f); // (0,1)
  float u2 = (float)(r2 >> 8) * (1.0f / 16777216.0f);
  float rad = 0.3f * sqrtf(-2.0f * __logf(u1));
  float ang = 6.28318530717958647f * u2;
  *n0 = rad * __cosf(ang);
  *n1 = rad * __sinf(ang);
}

// ---------------- Stage A: partial sums of the *input* rate buffers ----------------
__global__ void reduce_in_kernel(const float* __restrict__ day_rate,
                                 const float* __restrict__ night_rate,
                                 float* __restrict__ partials) {
  __shared__ float smem[8];
  int night = (blockIdx.x >= A_BLOCKS_PER_POP) ? 1 : 0;
  int pb = blockIdx.x - night * A_BLOCKS_PER_POP;
  const float* p = night ? night_rate : day_rate;
  float s = 0.0f;
  for (int i = pb * BLK + threadIdx.x; i < N_POP; i += A_BLOCKS_PER_POP * BLK) s += p[i];
  float bs = block_sum256(s, smem);
  if (threadIdx.x == 0) partials[blockIdx.x] = bs;
}

// ---------------- Stage A2: drive scalars ----------------
__global__ void prep_kernel(const float* __restrict__ apart,
                            const float* __restrict__ light,
                            const int* __restrict__ stepp,
                            float* __restrict__ scalars) {
  __shared__ float smem[8];
  int t = threadIdx.x;
  float sd = apart[t] + apart[t + 256] + apart[t + 512] + apart[t + 768];
  float sn = apart[1024 + t] + apart[1280 + t] + apart[1536 + t] + apart[1792 + t];
  float dtot = block_sum256(sd, smem);
  float ntot = block_sum256(sn, smem);
  if (t == 0) {
    float mean_d = dtot * (1.0f / (float)N_POP);
    float mean_n = ntot * (1.0f / (float)N_POP);
    int stp = stepp[0] + 1;
    float phase = (float)(stp % 2400) / 2400.0f;
    float ll = light[0];
    const float TWO_PI = 6.28318530717958647f;
    float day_drive   = fmaxf(0.0f, cosf(TWO_PI * (phase - 0.25f))) * 8.0f + ll * 3.0f;
    float night_drive = fmaxf(0.0f, cosf(TWO_PI * (phase - 0.75f))) * 8.0f - ll * 2.0f;
    scalars[0] = day_drive - mean_n * 5.0f + 2.0f;                 // I_day_base
    scalars[1] = fmaxf(0.0f, night_drive) - mean_d * 5.0f - 2.0f;  // I_night_base
    scalars[2] = phase;
  }
}

// ---------------- Stage B: 10 Izhikevich steps fully in registers ----------------
__global__ void __launch_bounds__(BLK)
izh_main_kernel(const float* __restrict__ dv_in, const float* __restrict__ du_in,
                const float* __restrict__ dr_in, const float* __restrict__ nv_in,
                const float* __restrict__ nu_in, const float* __restrict__ nr_in,
                const float* __restrict__ scalars, float* __restrict__ partials) {
  __shared__ float smem[8];
  const int nightB = (blockIdx.x >= MAIN_BLOCKS_PER_POP) ? 1 : 0;
  const int pb = blockIdx.x - nightB * MAIN_BLOCKS_PER_POP;
  const float* Vp = nightB ? nv_in : dv_in;
  const float* Up = nightB ? nu_in : du_in;
  const float* Rp = nightB ? nr_in : dr_in;
  const float Ibase = scalars[nightB];
  const uint32_t popbit = ((uint32_t)nightB) << 26;

  const int base = pb * (BLK * NPT) + threadIdx.x;
  float v[NPT], u[NPT], r[NPT];
#pragma unroll
  for (int k = 0; k < NPT; ++k) {
    int idx = base + k * BLK;
    v[k] = Vp[idx];
    u[k] = Up[idx];
    r[k] = Rp[idx];
  }
  // unique 32-bit RNG counters: idx < 2^22, step in bits 22..25, pop in bit 26
  const uint32_t id0 = (uint32_t)base;              // covers neurons k=0,1 (Box-Muller pair)
  const uint32_t id1 = (uint32_t)(base + 2 * BLK);  // covers neurons k=2,3

#pragma unroll
  for (int s = 0; s < N_INNER; ++s) {
    float nse[NPT];
    noise_pair(id0 | ((uint32_t)s << 22) | popbit, &nse[0], &nse[1]);
    noise_pair(id1 | ((uint32_t)s << 22) | popbit, &nse[2], &nse[3]);
#pragma unroll
    for (int k = 0; k < NPT; ++k) {
      float I   = Ibase + nse[k];
      float dvv = fmaf(v[k], fmaf(0.04f, v[k], 5.0f), 140.0f - u[k] + I);
      float v2  = v[k] + dvv;
      float u2  = fmaf(0.02f, fmaf(0.2f, v[k], -u[k]), u[k]);
      bool  spk = (v2 >= 30.0f);
      v[k] = spk ? -65.0f : v2;
      u[k] = spk ? (u2 + 8.0f) : u2;
      r[k] = fmaf(r[k], 0.9f, spk ? 0.1f : 0.0f);
    }
  }
  float s4 = (r[0] + r[1]) + (r[2] + r[3]);
  float bs = block_sum256(s4, smem);   // EXEC all-ones here (full blocks, uniform control flow)
  if (threadIdx.x == 0) partials[blockIdx.x] = bs;
}

// ---------------- Stage C: final means -> 5 outputs ----------------
__global__ void finalize_kernel(const float* __restrict__ mpart,
                                const float* __restrict__ scalars,
                                float* __restrict__ out) {
  __shared__ float smem[8];
  int t = threadIdx.x;
  float sd = 0.0f, sn = 0.0f;
  for (int i = t; i < MAIN_BLOCKS_PER_POP; i += BLK) {
    sd += mpart[i];
    sn += mpart[MAIN_BLOCKS_PER_POP + i];
  }
  float dtot = block_sum256(sd, smem);
  float ntot = block_sum256(sn, smem);
  if (t == 0) {
    float day_mean = dtot * (1.0f / (float)N_POP);
    float night_mean = ntot * (1.0f / (float)N_POP);
    float mel = night_mean / (day_mean + night_mean + 1e-8f);
    out[0] = scalars[2];                       // phase
    out[1] = mel;                              // melatonin
    out[2] = 0.3f + 0.7f * (1.0f - mel);       // activity_drive
    out[3] = day_mean;
    out[4] = night_mean;
  }
}

// ---------------- host-side launch ----------------
extern "C" void kernel_launch(void* const* d_in, const int* in_sizes, int n_in,
                              void* d_out, int out_size, void* d_ws, size_t ws_size,
                              hipStream_t stream) {
  const float* day_v      = (const float*)d_in[0];
  const float* day_u      = (const float*)d_in[1];
  const float* day_rate   = (const float*)d_in[2];
  const float* night_v    = (const float*)d_in[3];
  const float* night_u    = (const float*)d_in[4];
  const float* night_rate = (const float*)d_in[5];
  const float* light      = (const float*)d_in[6];
  const int*   step       = (const int*)d_in[7];
  float* ws   = (float*)d_ws;
  float* apart = ws + WS_APART_OFF;
  float* scal  = ws + WS_SCAL_OFF;
  float* mpart = ws + WS_MPART_OFF;
  float* out   = (float*)d_out;

  reduce_in_kernel<<<2 * A_BLOCKS_PER_POP, BLK, 0, stream>>>(day_rate, night_rate, apart);
  prep_kernel<<<1, BLK, 0, stream>>>(apart, light, step, scal);
  izh_main_kernel<<<2 * MAIN_BLOCKS_PER_POP, BLK, 0, stream>>>(
      day_v, day_u, day_rate, night_v, night_u, night_rate, scal, mpart);
  finalize_kernel<<<1, BLK, 0, stream>>>(mpart, scal, out);
}